// NCC_51969104282241
// MI455X (gfx1250) — compile-verified
//
#include <hip/hip_runtime.h>

#define DIM_X 160
#define DIM_Y 192
#define DIM_Z 160
#define NVOX  (DIM_X * DIM_Y * DIM_Z)   // 4,915,200
#define NLINES (DIM_Y * DIM_Z)          // 30,720 x-lines; also z-stride in elements
#define HALF 10
#define NWIN_F 9261.0f                  // 21^3

typedef float v2f __attribute__((ext_vector_type(2)));
typedef float v8f __attribute__((ext_vector_type(8)));
typedef int   vs4i __attribute__((vector_size(16)));   // GCC-style vector, per builtin proto

// ---- CDNA5 async global->LDS path (ASYNCcnt), with safe fallback ----------
#if defined(__has_builtin)
#if __has_builtin(__builtin_amdgcn_global_load_async_to_lds_b32) && \
    __has_builtin(__builtin_amdgcn_global_load_async_to_lds_b128) && \
    __has_builtin(__builtin_amdgcn_s_wait_asynccnt)
#define HAVE_ASYNC_LDS 1
#endif
#endif
#ifndef HAVE_ASYNC_LDS
#define HAVE_ASYNC_LDS 0
#endif

__device__ __forceinline__ void async_g2l_b32(const float* g, float* l) {
#if HAVE_ASYNC_LDS
  __builtin_amdgcn_global_load_async_to_lds_b32(
      (__attribute__((address_space(1))) int*)const_cast<float*>(g),
      (__attribute__((address_space(3))) int*)l, 0, 0);
#else
  *l = *g;
#endif
}

__device__ __forceinline__ void async_g2l_b128(const float* g, float* l) {
#if HAVE_ASYNC_LDS
  __builtin_amdgcn_global_load_async_to_lds_b128(
      (__attribute__((address_space(1))) vs4i*)const_cast<float*>(g),
      (__attribute__((address_space(3))) vs4i*)l, 0, 0);
#else
  l[0] = g[0]; l[1] = g[1]; l[2] = g[2]; l[3] = g[3];
#endif
}

__device__ __forceinline__ void async_wait_all() {
#if HAVE_ASYNC_LDS
  __builtin_amdgcn_s_wait_asynccnt(0);
#endif
}

// --------------------------------------------------------------------------
// Pass 1: pointwise products + 21-tap box sum along x via V_WMMA_F32_16X16X4_F32
// Out[m,n] = sum_{k=0}^{35} A[m,k]*S[k,n],  A[m,k]=in[line m, x_t-10+k],
// S[k,n] = 1 iff n <= k <= n+20   (zero-padded at volume edges)
// Block: 128 threads = 4 waves; each wave owns 16 lines; block owns 64 lines.
// --------------------------------------------------------------------------
__global__ __launch_bounds__(128) void ncc_pass_x(const float* __restrict__ I,
                                                  const float* __restrict__ J,
                                                  float* __restrict__ ws) {
  __shared__ float sI[64 * 84];
  __shared__ float sJ[64 * 84];

  const int tid  = threadIdx.x;
  const int wave = tid >> 5;
  const int lane = tid & 31;
  const int half = lane >> 4;   // 0: lanes 0-15, 1: lanes 16-31
  const int m16  = lane & 15;
  const int line0 = blockIdx.x * 64;

  // Banded selection matrix fragments (constant per lane): B[k, n], k = 4c + 2*half + r
  v2f bfrag[9];
  for (int c = 0; c < 9; ++c) {
    int k0 = 4 * c + 2 * half;
    bfrag[c].x = (k0     >= m16 && k0     <= m16 + 20) ? 1.0f : 0.0f;
    bfrag[c].y = (k0 + 1 >= m16 && k0 + 1 <= m16 + 20) ? 1.0f : 0.0f;
  }

  float* wsI  = ws + 0L * NVOX;
  float* wsJ  = ws + 1L * NVOX;
  float* wsI2 = ws + 2L * NVOX;
  float* wsJ2 = ws + 3L * NVOX;
  float* wsIJ = ws + 4L * NVOX;

  for (int x0 = 0; x0 < DIM_X; x0 += 64) {
    // Stage 64 lines x 84 floats (global x in [x0-10, x0+73]).
    // In-range elements: async global->LDS (no VGPR round-trip, ASYNCcnt).
    // Out-of-volume halo: zero via plain LDS store (DScnt).
    for (int idx = tid; idx < 64 * 84; idx += 128) {
      int l  = idx / 84;
      int j  = idx - l * 84;
      int gx = x0 - 10 + j;
      if (gx >= 0 && gx < DIM_X) {
        long g = (long)(line0 + l) * DIM_X + gx;
        async_g2l_b32(I + g, &sI[idx]);
        async_g2l_b32(J + g, &sJ[idx]);
      } else {
        sI[idx] = 0.0f;
        sJ[idx] = 0.0f;
      }
    }
    async_wait_all();
    __syncthreads();

    const int chunkW  = (DIM_X - x0) < 64 ? (DIM_X - x0) : 64;
    const int rowBase = (wave * 16 + m16) * 84;  // A-fragment LDS row (m = lane%16)

    for (int toff = 0; toff < chunkW; toff += 16) {
      v8f accI = {}, accJ = {}, accI2 = {}, accJ2 = {}, accIJ = {};
      for (int c = 0; c < 9; ++c) {
        int j = toff + 4 * c + 2 * half;        // k = 4c + 2*half + r
        v2f aI, aJ;
        aI.x = sI[rowBase + j]; aI.y = sI[rowBase + j + 1];
        aJ.x = sJ[rowBase + j]; aJ.y = sJ[rowBase + j + 1];
        v2f aI2 = aI * aI;
        v2f aJ2 = aJ * aJ;
        v2f aIJ = aI * aJ;
        accI  = __builtin_amdgcn_wmma_f32_16x16x4_f32(false, aI,  false, bfrag[c], (short)0, accI,  false, false);
        accJ  = __builtin_amdgcn_wmma_f32_16x16x4_f32(false, aJ,  false, bfrag[c], (short)0, accJ,  false, false);
        accI2 = __builtin_amdgcn_wmma_f32_16x16x4_f32(false, aI2, false, bfrag[c], (short)0, accI2, false, false);
        accJ2 = __builtin_amdgcn_wmma_f32_16x16x4_f32(false, aJ2, false, bfrag[c], (short)0, accJ2, false, false);
        accIJ = __builtin_amdgcn_wmma_f32_16x16x4_f32(false, aIJ, false, bfrag[c], (short)0, accIJ, false, false);
      }
      // D layout: VGPR v -> row m = v + 8*half, column n = lane%16
      for (int v = 0; v < 8; ++v) {
        int  mrow = v + 8 * half;
        long gi   = (long)(line0 + wave * 16 + mrow) * DIM_X + (x0 + toff + m16);
        wsI[gi]  = accI[v];
        wsJ[gi]  = accJ[v];
        wsI2[gi] = accI2[v];
        wsJ2[gi] = accJ2[v];
        wsIJ[gi] = accIJ[v];
      }
    }
    __syncthreads();
  }
}

// --------------------------------------------------------------------------
// Pass 2: 21-tap box sum along y, in place. One block = one z-slice, 32-wide
// x-slab; loops over the 5 channels. Tile staged via async b128 (rows are
// 16B-aligned); in-place global writeback is safe since the block only
// rewrites what it staged.
// --------------------------------------------------------------------------
__global__ __launch_bounds__(256) void ncc_pass_y(float* __restrict__ ws) {
  __shared__ float t[DIM_Y * 32];
  const int z  = blockIdx.x / 5;
  const int xb = blockIdx.x % 5;
  const int x0 = xb * 32;

  for (int ch = 0; ch < 5; ++ch) {
    float* base = ws + (long)ch * NVOX + (long)z * DIM_Y * DIM_X + x0;
    for (int q = threadIdx.x; q < DIM_Y * 8; q += 256) {  // 8 x-quads per row
      int y  = q >> 3;
      int x4 = (q & 7) * 4;
      async_g2l_b128(base + y * DIM_X + x4, &t[y * 32 + x4]);
    }
    async_wait_all();
    __syncthreads();
    for (int idx = threadIdx.x; idx < DIM_Y * 32; idx += 256) {
      int y = idx >> 5, x = idx & 31;
      int dlo = (-HALF > -y) ? -HALF : -y;
      int dhi = (HALF < DIM_Y - 1 - y) ? HALF : DIM_Y - 1 - y;
      float s = 0.0f;
      for (int d = dlo; d <= dhi; ++d) s += t[(y + d) * 32 + x];
      base[y * DIM_X + x] = s;
    }
    __syncthreads();
  }
}

// --------------------------------------------------------------------------
// Pass 3: sliding-window 21-tap sum along z + cc + reduction.
// One thread per (x,y) column; re-reads of the leaving plane are guaranteed
// L2 hits (192 MB L2 holds the whole 98 MB intermediate).
// --------------------------------------------------------------------------
__global__ __launch_bounds__(256) void ncc_pass_z_reduce(const float* __restrict__ ws,
                                                         double* __restrict__ acc) {
  const int col = blockIdx.x * 256 + threadIdx.x;  // 0..NLINES-1, x contiguous
  const float* p0 = ws + 0L * NVOX + col;  // I_sum
  const float* p1 = ws + 1L * NVOX + col;  // J_sum
  const float* p2 = ws + 2L * NVOX + col;  // I2_sum
  const float* p3 = ws + 3L * NVOX + col;  // J2_sum
  const float* p4 = ws + 4L * NVOX + col;  // IJ_sum

  float sI = 0.f, sJ = 0.f, sI2 = 0.f, sJ2 = 0.f, sIJ = 0.f;
  for (int z = 0; z < HALF; ++z) {
    int o = z * NLINES;
    sI += p0[o]; sJ += p1[o]; sI2 += p2[o]; sJ2 += p3[o]; sIJ += p4[o];
  }

  float local = 0.0f;
  for (int z = 0; z < DIM_Z; ++z) {
    if (z + HALF < DIM_Z) {
      int o = (z + HALF) * NLINES;
      sI += p0[o]; sJ += p1[o]; sI2 += p2[o]; sJ2 += p3[o]; sIJ += p4[o];
    }
    // mirror the reference's float32 formula exactly
    float u_I   = sI / NWIN_F;
    float u_J   = sJ / NWIN_F;
    float cross = sIJ - u_J * sI - u_I * sJ + u_I * u_J * NWIN_F;
    float I_var = sI2 - 2.0f * u_I * sI + u_I * u_I * NWIN_F;
    float J_var = sJ2 - 2.0f * u_J * sJ + u_J * u_J * NWIN_F;
    float cc    = cross * cross / (I_var * J_var + 1e-5f);
    local += cc;
    if (z >= HALF) {
      int o = (z - HALF) * NLINES;
      sI -= p0[o]; sJ -= p1[o]; sI2 -= p2[o]; sJ2 -= p3[o]; sIJ -= p4[o];
    }
  }

  __shared__ float red[256];
  red[threadIdx.x] = local;
  __syncthreads();
  for (int s = 128; s > 0; s >>= 1) {
    if (threadIdx.x < s) red[threadIdx.x] += red[threadIdx.x + s];
    __syncthreads();
  }
  if (threadIdx.x == 0) atomicAdd(acc, (double)red[0]);  // global_atomic_add_f64
}

__global__ void ncc_init(double* acc) { *acc = 0.0; }

__global__ void ncc_final(const double* __restrict__ acc, float* __restrict__ out) {
  out[0] = 1.0f - (float)(*acc / (double)NVOX);
}

// --------------------------------------------------------------------------
// d_ws layout: [0,8)   double accumulator
//              [256, 256 + 5*NVOX*4)  five f32 channels (~98.3 MB)
// --------------------------------------------------------------------------
extern "C" void kernel_launch(void* const* d_in, const int* in_sizes, int n_in,
                              void* d_out, int out_size, void* d_ws, size_t ws_size,
                              hipStream_t stream) {
  (void)in_sizes; (void)n_in; (void)out_size; (void)ws_size;
  const float* I = (const float*)d_in[0];
  const float* J = (const float*)d_in[1];
  double* acc  = (double*)d_ws;
  float* chans = (float*)((char*)d_ws + 256);

  ncc_init<<<1, 1, 0, stream>>>(acc);
  ncc_pass_x<<<NLINES / 64, 128, 0, stream>>>(I, J, chans);
  ncc_pass_y<<<DIM_Z * 5, 256, 0, stream>>>(chans);
  ncc_pass_z_reduce<<<NLINES / 256, 256, 0, stream>>>(chans, acc);
  ncc_final<<<1, 1, 0, stream>>>(acc, (float*)d_out);
}